// PDELoss_22883585753780
// MI455X (gfx1250) — compile-verified
//
#include <hip/hip_runtime.h>

// ---------------------------------------------------------------------------
// Fused GS-operator PDE loss for MI455X (gfx1250, wave32).
//   pass 1: per-block 16x64 output tile; pred + RR tiles staged in LDS with
//           alignment-safe float4 (b128) global loads; GS_ope halo tile in
//           LDS; Gaussian smooth + squared error; wave reduction via
//           V_WMMA_F32_16X16X4_F32; block partial -> d_ws
//   pass 2: deterministic reduction of block partials -> mean -> d_out[0]
// Memory-bound: ~190 MB mandatory traffic -> ~8.2 us floor at 23.3 TB/s.
// b128 loads cut VMEM request count 4x vs b32 at identical coalescing.
// ---------------------------------------------------------------------------

typedef __attribute__((ext_vector_type(2))) float v2f;
typedef __attribute__((ext_vector_type(8))) float v8f;

#define IMG_H   256
#define IMG_W   256
#define OUT_DIM 254
#define TX      64
#define TY      16
// pred tile: rows [i0-1, i0+18] (20), cols [j0-4, j0+67] (72 = 18 float4)
#define PT_H    20
#define PT_VW   18                 // float4s per pred row
// RR tile: rows [i0, i0+17] (18), cols [j0, j0+67] (68 -> 17 float4)
#define RT_H    18
#define RT_VW   17                 // float4s per RR row
#define ST_H    18                 // GS_ope halo tile rows
#define ST_W    66                 // GS_ope halo tile cols
#define LSTRIDE 72                 // LDS row stride (72 % 64 = 8 -> rotates banks)

// Exact wave32 sum via one fp32 WMMA.
// A (16x4) = each lane's partial duplicated into both A VGPRs: every lane
// value occupies exactly 2 of the 64 A slots. B = uniform 0.5, C = 0.
// => D[m,n] = 0.5 * rowsum(A) (independent of n since B is uniform), so
// summing all 16 M-rows of D gives sum(partials). Lanes 0-15 hold M=0..7 in
// c[0..7], lanes 16-31 hold M=8..15 -> lane0 + lane16 of the per-lane
// 8-element sum is the full wave sum. Requires EXEC all-ones: call from
// full, non-divergent waves only.
__device__ __forceinline__ float wave_sum_wmma(float acc) {
    v2f a; a.x = acc;  a.y = acc;
    v2f b; b.x = 0.5f; b.y = 0.5f;
    v8f c = {};
    c = __builtin_amdgcn_wmma_f32_16x16x4_f32(false, a, false, b,
                                              (short)0, c, false, false);
    float s = c[0] + c[1] + c[2] + c[3] + c[4] + c[5] + c[6] + c[7];
    return __shfl(s, 0, 32) + __shfl(s, 16, 32);
}

__global__ __launch_bounds__(256) void gs_loss_tile_kernel(
    const float* __restrict__ pred, const float* __restrict__ rhs,
    const float* __restrict__ Lk,   const float* __restrict__ Dk,
    const float* __restrict__ RR,   const float* __restrict__ ZZ,
    const float* __restrict__ Gk,   float* __restrict__ partials)
{
    __shared__ float PT[PT_H * LSTRIDE];   // pred tile (origin i0-1, j0-4)
    __shared__ float RT[RT_H * LSTRIDE];   // RR tile   (origin i0,   j0  )
    __shared__ float ST[ST_H * LSTRIDE];   // GS_ope halo tile (i0-1, j0-1)
    __shared__ float wred[8];

    const int b   = blockIdx.z;
    const int i0  = blockIdx.y * TY;       // output-tile origin (rows)
    const int j0  = blockIdx.x * TX;       // output-tile origin (cols)
    const int tid = threadIdx.x;

    const float* predB = pred + (size_t)b * IMG_H * IMG_W;
    const float* RRB   = RR   + (size_t)b * IMG_H * IMG_W;
    const float* ZZB   = ZZ   + (size_t)b * IMG_H * IMG_W;
    const float* rhsB  = rhs  + (size_t)b * OUT_DIM * OUT_DIM;

    // Per-batch uniform scalars (uniform addresses -> scalar loads).
    float L[9], D[9], G[9];
#pragma unroll
    for (int t = 0; t < 9; ++t) {
        L[t] = Lk[b * 9 + t];
        D[t] = Dk[b * 9 + t];
        G[t] = Gk[t];
    }
    const float hr  = RRB[1 * IMG_W + 2] - RRB[1 * IMG_W + 1];
    const float hz  = ZZB[2 * IMG_W + 1] - ZZB[1 * IMG_W + 1];
    const float hr2 = hr * hr, hz2 = hz * hz;
    const float scale = (-2.0f * (hr2 + hz2)) / (hr2 * hz2);

    // ---- Phase A: stage pred + RR tiles with aligned float4 loads ----
    // j0 is a multiple of 64 and IMG_W a multiple of 4, so every float4
    // group is either fully inside [0, IMG_W) or fully in the zero-pad
    // region — no mixed vectors. Row validity checked per row.
    for (int idx = tid; idx < PT_H * PT_VW; idx += 256) {
        const int lr = idx / PT_VW, vi = idx % PT_VW;
        const int gr = i0 - 1 + lr;
        const int gc = j0 - 4 + 4 * vi;
        float4 v = make_float4(0.f, 0.f, 0.f, 0.f);
        if (gr >= 0 && gr < IMG_H && gc >= 0 && gc + 3 < IMG_W)
            v = *(const float4*)(predB + gr * IMG_W + gc);
        *(float4*)(PT + lr * LSTRIDE + 4 * vi) = v;
    }
    for (int idx = tid; idx < RT_H * RT_VW; idx += 256) {
        const int lr = idx / RT_VW, vi = idx % RT_VW;
        const int gr = i0 + lr;
        const int gc = j0 + 4 * vi;
        float4 v = make_float4(1.f, 1.f, 1.f, 1.f);   // benign for pad region
        if (gr < IMG_H && gc + 3 < IMG_W)
            v = *(const float4*)(RRB + gr * IMG_W + gc);
        *(float4*)(RT + lr * LSTRIDE + 4 * vi) = v;
    }
    __syncthreads();

    // ---- Phase B: GS_ope halo tile (zero-padded SAME border) ----
    // lhs = Lpsi - Dpsi_dr = conv(pred,L) + conv(pred,Dk)/RR (signs cancel);
    // GS_ope = lhs * scale.
    for (int idx = tid; idx < ST_H * ST_W; idx += 256) {
        const int lr = idx / ST_W, lc = idx % ST_W;
        const int p = i0 - 1 + lr, q = j0 - 1 + lc;   // GS_ope coords
        float s = 0.0f;
        if (p >= 0 && p < OUT_DIM && q >= 0 && q < OUT_DIM) {
            float lp = 0.0f, dp = 0.0f;
#pragma unroll
            for (int u = 0; u < 3; ++u)
#pragma unroll
                for (int w = 0; w < 3; ++w) {
                    // pred[p+u][q+w] -> PT local col (q+w) - (j0-4) = lc+w+3
                    const float x = PT[(lr + u) * LSTRIDE + (lc + w + 3)];
                    lp = fmaf(x, L[u * 3 + w], lp);
                    dp = fmaf(x, D[u * 3 + w], dp);
                }
            // RR[p+1][q+1] -> RT[(p+1)-i0][(q+1)-j0] = RT[lr][lc]
            const float rv = RT[lr * LSTRIDE + lc];
            s = (lp + dp / rv) * scale;
        }
        ST[lr * LSTRIDE + lc] = s;
    }
    __syncthreads();

    // ---- Phase C: Gaussian SAME smooth + squared error vs rhs ----
    float acc = 0.0f;
#pragma unroll
    for (int k = 0; k < (TY * TX) / 256; ++k) {
        const int idx = tid + k * 256;
        const int ty = idx / TX, tx = idx % TX;
        const int i = i0 + ty, j = j0 + tx;
        if (i < OUT_DIM && j < OUT_DIM) {
            float sm = 0.0f;
#pragma unroll
            for (int u = 0; u < 3; ++u)
#pragma unroll
                for (int w = 0; w < 3; ++w)
                    sm = fmaf(G[u * 3 + w],
                              ST[(ty + u) * LSTRIDE + (tx + w)], sm);
            const float d = sm - rhsB[i * OUT_DIM + j];
            acc = fmaf(d, d, acc);
        }
    }

    // ---- Block reduction: WMMA per wave (8 full waves -> EXEC all-ones) ----
    const float ws = wave_sum_wmma(acc);
    if ((tid & 31) == 0) wred[tid >> 5] = ws;
    __syncthreads();
    if (tid == 0) {
        float t = 0.0f;
#pragma unroll
        for (int wv = 0; wv < 8; ++wv) t += wred[wv];
        partials[(size_t)blockIdx.z * (gridDim.x * gridDim.y) +
                 blockIdx.y * gridDim.x + blockIdx.x] = t;
    }
}

__global__ __launch_bounds__(256) void gs_loss_final_kernel(
    const float* __restrict__ partials, int n,
    float* __restrict__ out, float inv_count)
{
    __shared__ float wred[8];
    const int tid = threadIdx.x;
    float acc = 0.0f;
    const int n4 = n >> 2;
    const float4* p4 = (const float4*)partials;
    for (int i = tid; i < n4; i += 256) {
        const float4 v = p4[i];
        acc += (v.x + v.y) + (v.z + v.w);
    }
    for (int i = (n4 << 2) + tid; i < n; i += 256)   // tail (n % 4)
        acc += partials[i];
    const float ws = wave_sum_wmma(acc);
    if ((tid & 31) == 0) wred[tid >> 5] = ws;
    __syncthreads();
    if (tid == 0) {
        float t = 0.0f;
#pragma unroll
        for (int wv = 0; wv < 8; ++wv) t += wred[wv];
        out[0] = t * inv_count;
    }
}

extern "C" void kernel_launch(void* const* d_in, const int* in_sizes, int n_in,
                              void* d_out, int out_size, void* d_ws, size_t ws_size,
                              hipStream_t stream) {
    const float* pred = (const float*)d_in[0];
    const float* rhs  = (const float*)d_in[1];
    const float* Lk   = (const float*)d_in[2];
    const float* Dk   = (const float*)d_in[3];
    const float* RR   = (const float*)d_in[4];
    const float* ZZ   = (const float*)d_in[5];
    const float* Gk   = (const float*)d_in[6];
    float* out       = (float*)d_out;
    float* partials  = (float*)d_ws;

    const int B = in_sizes[2] / 9;   // per-sample 3x3 kernels -> batch count

    dim3 grid((OUT_DIM + TX - 1) / TX,   // 4
              (OUT_DIM + TY - 1) / TY,   // 16
              B);                        // 256
    gs_loss_tile_kernel<<<grid, 256, 0, stream>>>(pred, rhs, Lk, Dk, RR, ZZ,
                                                  Gk, partials);

    const int nblk = grid.x * grid.y * grid.z;
    const float inv_count =
        (float)(1.0 / ((double)B * (double)OUT_DIM * (double)OUT_DIM));
    gs_loss_final_kernel<<<1, 256, 0, stream>>>(partials, nblk, out, inv_count);
}